// MambaEncoder_67886253081097
// MI455X (gfx1250) — compile-verified
//
#include <hip/hip_runtime.h>

#define D_MODEL 1024
#define D_STATE 16
#define D_CONV  4
#define D_INNER 2048
#define DT_RANK 64
#define BATCH   2
#define SEQ     1024
#define ROWS    (BATCH*SEQ)
#define XDBL_W  (DT_RANK + 2*D_STATE)   // 96
#define LDP     20                      // padded LDS row stride (floats): conflict-free

typedef __attribute__((ext_vector_type(2))) float v2f;
typedef __attribute__((ext_vector_type(4))) float v4f;
typedef __attribute__((ext_vector_type(8))) float v8f;
typedef __attribute__((ext_vector_type(4))) int   v4i;

#if __has_builtin(__builtin_amdgcn_global_load_async_to_lds_b128)
#define USE_ASYNC 1
#if __has_builtin(__builtin_amdgcn_s_wait_asynccnt)
#define WAIT_ASYNC(n) __builtin_amdgcn_s_wait_asynccnt(n)
#else
#define WAIT_ASYNC(n) asm volatile("s_wait_asynccnt %0" ::"n"(n) : "memory")
#endif
// signature (from clang diagnostic): (int4 as1* src, int4 as3* dst, imm off, imm cpol)
#define ASYNC_CP16(dst, src)                                          \
  __builtin_amdgcn_global_load_async_to_lds_b128(                     \
      (__attribute__((address_space(1))) v4i*)(src),                  \
      (__attribute__((address_space(3))) v4i*)(dst), 0, 0)
#else
#define USE_ASYNC 0
#endif

__device__ __forceinline__ float sigmoidf_(float x) { return 1.f / (1.f + __expf(-x)); }
__device__ __forceinline__ float softplusf_(float x) {
  return fmaxf(x, 0.f) + log1pf(__expf(-fabsf(x)));
}

// ---------------- LayerNorm over D_MODEL, one block per row ----------------
__global__ void ln_kernel(const float* __restrict__ src, const float* __restrict__ g,
                          const float* __restrict__ b, float* __restrict__ dst) {
  int row = blockIdx.x;
  const float* x = src + (size_t)row * D_MODEL;
  float s = 0.f, q = 0.f;
  for (int i = threadIdx.x; i < D_MODEL; i += blockDim.x) { float v = x[i]; s += v; q += v * v; }
  __shared__ float sA[8], sB[8], st[2];
  for (int o = 16; o > 0; o >>= 1) { s += __shfl_down(s, o, 32); q += __shfl_down(q, o, 32); }
  if ((threadIdx.x & 31) == 0) { sA[threadIdx.x >> 5] = s; sB[threadIdx.x >> 5] = q; }
  __syncthreads();
  if (threadIdx.x == 0) {
    float ts = 0.f, tq = 0.f;
    for (int i = 0; i < (int)(blockDim.x >> 5); ++i) { ts += sA[i]; tq += sB[i]; }
    float mu = ts / (float)D_MODEL;
    float var = tq / (float)D_MODEL - mu * mu;
    st[0] = mu; st[1] = rsqrtf(var + 1e-5f);
  }
  __syncthreads();
  float mu = st[0], r = st[1];
  float* y = dst + (size_t)row * D_MODEL;
  for (int i = threadIdx.x; i < D_MODEL; i += blockDim.x)
    y[i] = (x[i] - mu) * r * g[i] + b[i];
}

// ======== LDS-staged, double-buffered fp32 WMMA GEMM ========
// C[M,N] = A[M,K] * W[N,K]^T.  Block tile 128x128, BK=16, 256 threads (8 waves,
// 4x2 wave grid of 32x64 wave tiles).  A/B tiles staged to LDS via
// GLOBAL_LOAD_ASYNC_TO_LDS_B128 (ASYNCcnt-pipelined) when available,
// fragments fed to v_wmma_f32_16x16x4_f32 via ds_load_b64.
// EPI: 0=store, 1=accumulate Cin (residual), 2=softplus(v+bias[n]).
template<int EPI>
__global__ __launch_bounds__(256) void wmma_gemm_lds(
    const float* __restrict__ A, int lda,
    const float* __restrict__ W, int ldw,
    float* __restrict__ C, int ldc,
    const float* __restrict__ Cin,
    const float* __restrict__ bias,
    int M, int N, int K) {
  __shared__ __align__(16) float smem[2][2][128 * LDP];

  const int tid  = threadIdx.x;
  const int lane = tid & 31;
  const int w    = tid >> 5;
  const int waveM = w >> 1;            // 0..3
  const int waveN = w & 1;             // 0..1
  const int half = lane >> 4;          // K 0,1 / rows 0..7  vs  K 2,3 / rows 8..15
  const int l16  = lane & 15;
  const int koff = half << 1;

  const int nb = N / 128;
  const int bm = blockIdx.x / nb, bn = blockIdx.x % nb;
  const int m0 = bm * 128, n0 = bn * 128;

  // per-thread copy coordinates: rows r and r+64, one 16B chunk of the 16-wide K slab
  const int r  = tid >> 2;
  const int kc = (tid & 3) << 2;
  const float* gA  = A + (size_t)(m0 + r)      * lda + kc;
  const float* gA2 = A + (size_t)(m0 + r + 64) * lda + kc;
  const float* gB  = W + (size_t)(n0 + r)      * ldw + kc;
  const float* gB2 = W + (size_t)(n0 + r + 64) * ldw + kc;
  const int lo  = r * LDP + kc;
  const int lo2 = (r + 64) * LDP + kc;

  v8f acc[2][4] = {};
  const int NKB = K / 16;

  auto compute = [&](int buf) {
    const float* lA = &smem[buf][0][0];
    const float* lB = &smem[buf][1][0];
#pragma unroll
    for (int ks = 0; ks < 4; ++ks) {
      v2f af[2], bf[4];
#pragma unroll
      for (int ms = 0; ms < 2; ++ms)
        af[ms] = *(const v2f*)&lA[(waveM * 32 + ms * 16 + l16) * LDP + ks * 4 + koff];
#pragma unroll
      for (int ns = 0; ns < 4; ++ns)
        bf[ns] = *(const v2f*)&lB[(waveN * 64 + ns * 16 + l16) * LDP + ks * 4 + koff];
#pragma unroll
      for (int ms = 0; ms < 2; ++ms)
#pragma unroll
        for (int ns = 0; ns < 4; ++ns)
          acc[ms][ns] = __builtin_amdgcn_wmma_f32_16x16x4_f32(
              false, af[ms], false, bf[ns], (short)0, acc[ms][ns], false, false);
    }
  };

#if USE_ASYNC
  auto issue = [&](int buf, int kb) {
    size_t ko = (size_t)kb * 16;
    ASYNC_CP16(&smem[buf][0][lo],  gA  + ko);
    ASYNC_CP16(&smem[buf][0][lo2], gA2 + ko);
    ASYNC_CP16(&smem[buf][1][lo],  gB  + ko);
    ASYNC_CP16(&smem[buf][1][lo2], gB2 + ko);
  };
  issue(0, 0);
  for (int kb = 0; kb < NKB; ++kb) {
    int cur = kb & 1;
    if (kb + 1 < NKB) { issue(1 - cur, kb + 1); WAIT_ASYNC(4); }
    else              { WAIT_ASYNC(0); }
    __syncthreads();          // current stage resident for all waves
    compute(cur);
    __syncthreads();          // everyone done reading before buffer reuse
  }
#else
  struct Frags { v4f a0, a1, b0, b1; };
  auto gload = [&](int kb) {
    size_t ko = (size_t)kb * 16;
    Frags f;
    f.a0 = *(const v4f*)(gA + ko);  f.a1 = *(const v4f*)(gA2 + ko);
    f.b0 = *(const v4f*)(gB + ko);  f.b1 = *(const v4f*)(gB2 + ko);
    return f;
  };
  Frags fr = gload(0);
  for (int kb = 0; kb < NKB; ++kb) {
    int cur = kb & 1;
    *(v4f*)&smem[cur][0][lo]  = fr.a0;
    *(v4f*)&smem[cur][0][lo2] = fr.a1;
    *(v4f*)&smem[cur][1][lo]  = fr.b0;
    *(v4f*)&smem[cur][1][lo2] = fr.b1;
    __syncthreads();
    if (kb + 1 < NKB) fr = gload(kb + 1);   // next stage in flight during compute
    compute(cur);
    __syncthreads();
  }
#endif

#pragma unroll
  for (int ms = 0; ms < 2; ++ms)
#pragma unroll
    for (int i = 0; i < 8; ++i) {
      int mm = m0 + waveM * 32 + ms * 16 + (half << 3) + i;
#pragma unroll
      for (int ns = 0; ns < 4; ++ns) {
        int n = n0 + waveN * 64 + ns * 16 + l16;
        float v = acc[ms][ns][i];
        if (EPI == 1) v += Cin[(size_t)mm * ldc + n];
        if (EPI == 2) v = softplusf_(v + bias[n]);
        C[(size_t)mm * ldc + n] = v;
      }
    }
}

// ---------------- direct WMMA GEMM (small N; used for x_proj N=96) ----------------
template<int NSUB, int EPI>
__global__ void wmma_gemm(const float* __restrict__ A, int lda,
                          const float* __restrict__ W, int ldw,
                          float* __restrict__ C, int ldc,
                          const float* __restrict__ Cin,
                          const float* __restrict__ bias,
                          int M, int N, int K) {
  int lane  = threadIdx.x & 31;
  int wid   = (blockIdx.x * blockDim.x + threadIdx.x) >> 5;
  int tilesN = N / (16 * NSUB);
  int tilesM = M / 16;
  if (wid >= tilesM * tilesN) return;
  int tm = wid / tilesN, tn = wid % tilesN;
  int half = lane >> 4;
  int l16  = lane & 15;
  int koff = half << 1;
  int m    = tm * 16 + l16;
  int nbase = tn * 16 * NSUB;

  const float* Arow = A + (size_t)m * lda + koff;
  const float* Wp[NSUB];
#pragma unroll
  for (int s2 = 0; s2 < NSUB; ++s2)
    Wp[s2] = W + (size_t)(nbase + s2 * 16 + l16) * ldw + koff;

  v8f acc[NSUB] = {};
  for (int k0 = 0; k0 < K; k0 += 4) {
    v2f a = *(const v2f*)(Arow + k0);
#pragma unroll
    for (int s2 = 0; s2 < NSUB; ++s2) {
      v2f bb = *(const v2f*)(Wp[s2] + k0);
      acc[s2] = __builtin_amdgcn_wmma_f32_16x16x4_f32(
          false, a, false, bb, (short)0, acc[s2], false, false);
    }
  }

#pragma unroll
  for (int i = 0; i < 8; ++i) {
    int mm = tm * 16 + (half << 3) + i;
#pragma unroll
    for (int s2 = 0; s2 < NSUB; ++s2) {
      int n = nbase + s2 * 16 + l16;
      float v = acc[s2][i];
      if (EPI == 1) v += Cin[(size_t)mm * ldc + n];
      if (EPI == 2) v = softplusf_(v + bias[n]);
      C[(size_t)mm * ldc + n] = v;
    }
  }
}

// -------- causal depthwise conv (4 taps) over time + bias + SiLU --------
__global__ void conv_silu_kernel(const float* __restrict__ xz,
                                 const float* __restrict__ cw,
                                 const float* __restrict__ cb,
                                 float* __restrict__ xm) {
  int idx = blockIdx.x * blockDim.x + threadIdx.x;
  if (idx >= BATCH * SEQ * D_INNER) return;
  int d = idx & (D_INNER - 1);
  int l = (idx >> 11) & (SEQ - 1);
  int b = idx >> 21;
  float s = cb[d];
#pragma unroll
  for (int j = 0; j < D_CONV; ++j) {
    int ll = l - (D_CONV - 1) + j;
    if (ll >= 0)
      s += cw[d * D_CONV + j] * xz[((size_t)b * SEQ + ll) * (2 * D_INNER) + d];
  }
  xm[((size_t)b * SEQ + l) * D_INNER + d] = s * sigmoidf_(s);
}

// -------- selective scan: lane = (b,d,s); 16-lane shfl reduction over states;
//          fused D-skip and SiLU(z) gating; y written in-place over dt --------
__global__ void scan_kernel(float* __restrict__ dty,
                            const float* __restrict__ xdbl,
                            const float* __restrict__ xm,
                            const float* __restrict__ xz,
                            const float* __restrict__ A_log,
                            const float* __restrict__ Dskip) {
  int tid = blockIdx.x * blockDim.x + threadIdx.x;
  int s = tid & 15;
  int d = (tid >> 4) & (D_INNER - 1);
  int b = tid >> 15;
  if (b >= BATCH) return;
  float Aval = -__expf(A_log[d * D_STATE + s]);
  float Dv = Dskip[d];
  float h = 0.f;
  for (int l = 0; l < SEQ; ++l) {
    size_t row = (size_t)b * SEQ + l;
    float dtv = dty[row * D_INNER + d];
    float xv  = xm[row * D_INNER + d];
    float bv  = xdbl[row * XDBL_W + DT_RANK + s];
    float cv  = xdbl[row * XDBL_W + DT_RANK + D_STATE + s];
    h = __expf(dtv * Aval) * h + (dtv * bv) * xv;
    float acc = h * cv;
    acc += __shfl_xor(acc, 1, 16);
    acc += __shfl_xor(acc, 2, 16);
    acc += __shfl_xor(acc, 4, 16);
    acc += __shfl_xor(acc, 8, 16);
    if (s == 0) {
      float zv = xz[row * (2 * D_INNER) + D_INNER + d];
      dty[row * D_INNER + d] = (acc + xv * Dv) * (zv * sigmoidf_(zv));
    }
  }
}

extern "C" void kernel_launch(void* const* d_in, const int* in_sizes, int n_in,
                              void* d_out, int out_size, void* d_ws, size_t ws_size,
                              hipStream_t stream) {
  const float* x_in  = (const float*)d_in[0];
  const float* in_w  = (const float*)d_in[1];
  const float* cw    = (const float*)d_in[2];
  const float* cb    = (const float*)d_in[3];
  const float* xp_w  = (const float*)d_in[4];
  const float* dt_w  = (const float*)d_in[5];
  const float* dt_b  = (const float*)d_in[6];
  const float* A_log = (const float*)d_in[7];
  const float* Dskip = (const float*)d_in[8];
  const float* out_w = (const float*)d_in[9];
  const float* ln_g  = (const float*)d_in[10];
  const float* ln_b  = (const float*)d_in[11];
  float* out = (float*)d_out;

  float* ws    = (float*)d_ws;
  float* x_res = ws;                                   // [ROWS, D_MODEL]
  float* ln_x  = x_res + (size_t)ROWS * D_MODEL;       // [ROWS, D_MODEL]
  float* xz    = ln_x  + (size_t)ROWS * D_MODEL;       // [ROWS, 2*D_INNER]
  float* xm    = xz    + (size_t)ROWS * 2 * D_INNER;   // [ROWS, D_INNER]
  float* xdbl  = xm    + (size_t)ROWS * D_INNER;       // [ROWS, 96]
  float* dty   = xdbl  + (size_t)ROWS * XDBL_W;        // [ROWS, D_INNER] (dt, then y)

  (void)hipMemcpyAsync(x_res, x_in, sizeof(float) * (size_t)ROWS * D_MODEL,
                       hipMemcpyDeviceToDevice, stream);

  for (int i = 0; i < 4; ++i) {
    ln_kernel<<<ROWS, 256, 0, stream>>>(x_res, ln_g, ln_b, ln_x);

    // xz = ln_x @ in_w^T   [2048x1024] x [4096x1024]^T   -> 512 blocks
    wmma_gemm_lds<0><<<(ROWS / 128) * (2 * D_INNER / 128), 256, 0, stream>>>(
        ln_x, D_MODEL, in_w + (size_t)i * 2 * D_INNER * D_MODEL, D_MODEL,
        xz, 2 * D_INNER, nullptr, nullptr, ROWS, 2 * D_INNER, D_MODEL);

    conv_silu_kernel<<<(BATCH * SEQ * D_INNER) / 256, 256, 0, stream>>>(
        xz, cw + (size_t)i * D_INNER * D_CONV, cb + (size_t)i * D_INNER, xm);

    // xdbl = xm @ xp_w^T   [2048x2048] x [96x2048]^T (N=96 -> direct kernel)
    wmma_gemm<1, 0><<<((ROWS / 16) * (XDBL_W / 16) * 32 + 255) / 256, 256, 0, stream>>>(
        xm, D_INNER, xp_w + (size_t)i * XDBL_W * D_INNER, D_INNER,
        xdbl, XDBL_W, nullptr, nullptr, ROWS, XDBL_W, D_INNER);

    // dt = softplus(xdbl[:, :64] @ dt_w^T + dt_b)   K=64 -> 4 LDS stages
    wmma_gemm_lds<2><<<(ROWS / 128) * (D_INNER / 128), 256, 0, stream>>>(
        xdbl, XDBL_W, dt_w + (size_t)i * D_INNER * DT_RANK, DT_RANK,
        dty, D_INNER, nullptr, dt_b + (size_t)i * D_INNER, ROWS, D_INNER, DT_RANK);

    scan_kernel<<<(BATCH * D_INNER * D_STATE) / 256, 256, 0, stream>>>(
        dty, xdbl, xm, xz, A_log + (size_t)i * D_INNER * D_STATE,
        Dskip + (size_t)i * D_INNER);

    // x_res += y @ out_w^T  (fused residual accumulate)
    wmma_gemm_lds<1><<<(ROWS / 128) * (D_MODEL / 128), 256, 0, stream>>>(
        dty, D_INNER, out_w + (size_t)i * D_MODEL * D_INNER, D_INNER,
        x_res, D_MODEL, x_res, nullptr, ROWS, D_MODEL, D_INNER);
  }

  ln_kernel<<<ROWS, 256, 0, stream>>>(x_res, ln_g, ln_b, out);
}